// PairwiseConv_48430051230235
// MI455X (gfx1250) — compile-verified
//
#include <hip/hip_runtime.h>
#include <hip/hip_bf16.h>

typedef __attribute__((ext_vector_type(2))) float v2f;
typedef __attribute__((ext_vector_type(8))) float v8f;

#define B_       8
#define W_       256
#define EDGE_C   64
#define NODE_C   128
#define HID_     64
#define NT_      8
#define WW_      (W_ * W_)            // 65536
#define NPIX_    (B_ * WW_)           // 524288 elements per channel for BN stats
#define BN_EPS_  1e-5f
#define NPART_   1024                 // kernel-2 grid (partial-sum slots)

// ---- workspace layout (in floats) ----
#define X_OFF    ((size_t)0)
#define X_SZ     ((size_t)B_ * HID_ * WW_)          // 134.2 MB
#define NF_OFF   (X_OFF + X_SZ)
#define NN_SZ    ((size_t)B_ * HID_ * W_)
#define NG_OFF   (NF_OFF + NN_SZ)
#define PSUM_OFF (NG_OFF + NN_SZ)
#define PSQ_OFF  (PSUM_OFF + (size_t)NPART_ * HID_)
#define SCSH_OFF (PSQ_OFF + (size_t)NPART_ * HID_)

// Async global->LDS copy (CDNA5): LDS dest offset = low 32 bits of flat
// shared pointer (LDS aperture low bits are zero per ISA aperture rules).
__device__ __forceinline__ void async_g2l_b32(void* ldsDst, const float* gsrc) {
  unsigned ldsOff = (unsigned)(uintptr_t)ldsDst;
  asm volatile("global_load_async_to_lds_b32 %0, %1, off"
               :
               : "v"(ldsOff), "v"(gsrc)
               : "memory");
}
__device__ __forceinline__ void wait_asynccnt0() {
  asm volatile("s_wait_asynccnt 0x0" ::: "memory");
}

// ---------------------------------------------------------------------------
// Kernel 1: nf[b,o,j] = wf@node, ng[b,o,i] = wg@node  (tiny: 67 MFLOP)
// ---------------------------------------------------------------------------
__global__ __launch_bounds__(256) void k1_nodeproj(
    const float* __restrict__ node, const float* __restrict__ w_aff,
    float* __restrict__ nf, float* __restrict__ ng) {
  int idx = blockIdx.x * 256 + threadIdx.x;   // (b,o,j) flattened, 131072 total
  int j = idx & 255;
  int o = (idx >> 8) & 63;
  int b = idx >> 14;
  const float* np = node + (size_t)b * NODE_C * W_ + j;
  const float* wf = w_aff + (size_t)o * 320;        // row of (HID, 320)
  const float* wg = wf + NODE_C;
  float f = 0.f, g = 0.f;
#pragma unroll 8
  for (int c = 0; c < NODE_C; ++c) {
    float nv = np[(size_t)c * W_];
    f = fmaf(wf[c], nv, f);
    g = fmaf(wg[c], nv, g);
  }
  nf[idx] = f;
  ng[idx] = g;
}

// ---------------------------------------------------------------------------
// Kernel 2: x[b,o,i,j] = sum_c we[o,c]*edge[b,c,i,j] + nf[b,o,j] + ng[b,o,i]
//   64 x V_WMMA_F32_16X16X4_F32 per 16-pixel tile; B fragments software-
//   pipelined (prefetch distance 2) so HBM latency overlaps WMMA issue.
// ---------------------------------------------------------------------------
__global__ __launch_bounds__(256) void k2_edgeproj(
    const float* __restrict__ edge, const float* __restrict__ w_aff,
    const float* __restrict__ nf, const float* __restrict__ ng,
    float* __restrict__ x, float* __restrict__ psum, float* __restrict__ psq) {
  __shared__ float sWe[HID_ * EDGE_C];   // we[o][c], 16 KB
  __shared__ float wred[8 * HID_];       // per-wave channel sums
  __shared__ float wredq[8 * HID_];

  int tid = threadIdx.x;
  for (int idx = tid; idx < HID_ * EDGE_C; idx += 256) {
    int o = idx >> 6, c = idx & 63;
    sWe[idx] = w_aff[(size_t)o * 320 + 2 * NODE_C + c];   // we = w_aff[:,256:320]
  }
  __syncthreads();
  const v2f* sWeV = (const v2f*)sWe;

  int lane = tid & 31, wv = tid >> 5;
  int ln = lane & 15, half = lane >> 4;   // wave32: low/high half-wave
  int waveG = blockIdx.x * 8 + wv;
  int nwaves = gridDim.x * 8;             // 8192 waves, 4 tiles each

  v8f ssum[4] = {};   // per-lane running BN sums, rows o = 16m + r + 8*half
  v8f ssq[4]  = {};

  const int NTILES = B_ * W_ * (W_ / 16);  // 32768 (b, i, j-tile)
  for (int t = waveG; t < NTILES; t += nwaves) {   // wave-uniform => EXEC all-1s
    int b = t >> 12;
    int rem = t & 4095;
    int iRow = rem >> 4;
    int j0 = (rem & 15) << 4;

    const float* eb = edge + (size_t)b * EDGE_C * WW_ + (size_t)iRow * W_ + j0 + ln;
    v8f acc[4] = {};

    // B fragment (4x16): lanes 0-15 carry K=4k,4k+1; lanes 16-31 K=4k+2,4k+3
    v2f bq0, bq1;
    {
      int c0 = 2 * half;
      bq0.x = eb[(size_t)c0 * WW_];
      bq0.y = eb[(size_t)(c0 + 1) * WW_];
      int c1 = 4 + 2 * half;
      bq1.x = eb[(size_t)c1 * WW_];
      bq1.y = eb[(size_t)(c1 + 1) * WW_];
    }
#pragma unroll
    for (int k = 0; k < 16; ++k) {        // K = 64 channels, 4 per WMMA
      v2f bnext = bq1;
      if (k < 14) {                       // prefetch k+2 (compile-time cond)
        int cn = 4 * (k + 2) + 2 * half;
        bnext.x = eb[(size_t)cn * WW_];
        bnext.y = eb[(size_t)(cn + 1) * WW_];
      }
      int ac = 2 * k + half;              // (col)/2 in v2f units
      v2f A0 = sWeV[(size_t)(0 * 16 + ln) * 32 + ac];
      v2f A1 = sWeV[(size_t)(1 * 16 + ln) * 32 + ac];
      v2f A2 = sWeV[(size_t)(2 * 16 + ln) * 32 + ac];
      v2f A3 = sWeV[(size_t)(3 * 16 + ln) * 32 + ac];
      acc[0] = __builtin_amdgcn_wmma_f32_16x16x4_f32(false, A0, false, bq0, (short)0, acc[0], false, false);
      acc[1] = __builtin_amdgcn_wmma_f32_16x16x4_f32(false, A1, false, bq0, (short)0, acc[1], false, false);
      acc[2] = __builtin_amdgcn_wmma_f32_16x16x4_f32(false, A2, false, bq0, (short)0, acc[2], false, false);
      acc[3] = __builtin_amdgcn_wmma_f32_16x16x4_f32(false, A3, false, bq0, (short)0, acc[3], false, false);
      bq0 = bq1;
      bq1 = bnext;
    }

    // epilogue: + nf + ng, BN stats, store x
    float* xb = x + (size_t)b * HID_ * WW_ + (size_t)iRow * W_ + j0 + ln;
    const float* nfb = nf + (size_t)b * HID_ * W_;
    const float* ngb = ng + (size_t)b * HID_ * W_;
#pragma unroll
    for (int m = 0; m < 4; ++m) {
#pragma unroll
      for (int r = 0; r < 8; ++r) {
        int o = 16 * m + r + 8 * half;    // C/D layout: V r, lanes16-31 = M+8
        float v = acc[m][r] + nfb[(size_t)o * W_ + j0 + ln]
                            + ngb[(size_t)o * W_ + iRow];
        ssum[m][r] += v;
        ssq[m][r] = fmaf(v, v, ssq[m][r]);
        xb[(size_t)o * WW_] = v;
      }
    }
  }

  // deterministic reduction: lanes -> wave slot -> block partial
#pragma unroll
  for (int m = 0; m < 4; ++m) {
#pragma unroll
    for (int r = 0; r < 8; ++r) {
      float s = ssum[m][r], q = ssq[m][r];
#pragma unroll
      for (int d = 1; d < 16; d <<= 1) {  // reduce within each 16-lane half
        s += __shfl_xor(s, d, 32);
        q += __shfl_xor(q, d, 32);
      }
      if (ln == 0) {                      // lane 0 and lane 16 each own one o
        int o = 16 * m + r + 8 * half;
        wred[wv * HID_ + o]  = s;
        wredq[wv * HID_ + o] = q;
      }
    }
  }
  __syncthreads();
  if (tid < HID_) {
    float s = 0.f, q = 0.f;
#pragma unroll
    for (int w = 0; w < 8; ++w) { s += wred[w * HID_ + tid]; q += wredq[w * HID_ + tid]; }
    psum[(size_t)blockIdx.x * HID_ + tid] = s;
    psq[(size_t)blockIdx.x * HID_ + tid]  = q;
  }
}

// ---------------------------------------------------------------------------
// Kernel 3: finalize BN -> per-channel affine  scale=a, shift=beta - a*mean
// ---------------------------------------------------------------------------
__global__ __launch_bounds__(64) void k3_stats(
    const float* __restrict__ psum, const float* __restrict__ psq,
    const float* __restrict__ gamma, const float* __restrict__ beta,
    float* __restrict__ scsh) {
  int o = threadIdx.x;
  float s = 0.f, q = 0.f;
  for (int p = 0; p < NPART_; ++p) {
    s += psum[(size_t)p * HID_ + o];
    q += psq[(size_t)p * HID_ + o];
  }
  float inv = 1.0f / (float)NPIX_;
  float mean = s * inv;
  float var = fmaf(q, inv, -mean * mean);
  float a = gamma[o] * rsqrtf(var + BN_EPS_);
  scsh[o] = a;
  scsh[HID_ + o] = fmaf(-a, mean, beta[o]);
}

// ---------------------------------------------------------------------------
// Kernel 4: out = w_out @ (relu(aff(x)) + relu(aff(x))^T) + b_out
//   Symmetric in (i,j): only tile pairs ti<=tj computed; mirror written via
//   LDS transpose. Transposed-tile staging uses CDNA5 async global->LDS.
// ---------------------------------------------------------------------------
__global__ __launch_bounds__(256) void k4_out(
    const float* __restrict__ x, const float* __restrict__ w_out,
    const float* __restrict__ b_out, const float* __restrict__ scsh,
    float* __restrict__ out) {
  __shared__ float sW[NT_ * HID_];        // 8x64
  __shared__ float sA[HID_], sC[HID_];
  __shared__ float sB[NT_];
  __shared__ float sT[8][16][17];         // transposed-tile staging (pad vs banks)

  int tid = threadIdx.x;
  int tx = tid & 15, ty = tid >> 4;
  int b = blockIdx.y;
  // triangular decode of tile pair (ti <= tj), 136 pairs
  int p = blockIdx.x, ti = 0;
  while (p >= (16 - ti)) { p -= (16 - ti); ++ti; }
  int tj = ti + p;
  int i0 = ti << 4, j0 = tj << 4;

  for (int idx = tid; idx < NT_ * HID_; idx += 256) sW[idx] = w_out[idx];
  if (tid < HID_) { sA[tid] = scsh[tid]; sC[tid] = scsh[HID_ + tid]; }
  if (tid < NT_)  sB[tid] = b_out[tid];
  __syncthreads();

  size_t xb = (size_t)b * HID_ * WW_;
  float accT[NT_] = {};
  for (int c0 = 0; c0 < HID_; c0 += 8) {
    __syncthreads();                      // protect sT against previous reads
#pragma unroll
    for (int cc = 0; cc < 8; ++cc)        // async copy mirror-tile rows to LDS
      async_g2l_b32(&sT[cc][ty][tx],
                    &x[xb + (size_t)(c0 + cc) * WW_ + (size_t)(j0 + ty) * W_ + i0 + tx]);
    wait_asynccnt0();                     // LDS writes landed for this wave
    __syncthreads();                      // ... and for all waves in block
#pragma unroll
    for (int cc = 0; cc < 8; ++cc) {
      int c = c0 + cc;
      float a = sA[c], sh = sC[c];
      float d = x[xb + (size_t)c * WW_ + (size_t)(i0 + ty) * W_ + j0 + tx];
      float v1 = fmaxf(fmaf(a, d, sh), 0.f);
      float v2 = fmaxf(fmaf(a, sT[cc][tx][ty], sh), 0.f);  // x[b,c,j,i]
      float s = v1 + v2;
#pragma unroll
      for (int t = 0; t < NT_; ++t) accT[t] = fmaf(sW[t * HID_ + c], s, accT[t]);
    }
  }

  size_t ob = (size_t)b * NT_ * WW_;
#pragma unroll
  for (int t = 0; t < NT_; ++t)
    out[ob + (size_t)t * WW_ + (size_t)(i0 + ty) * W_ + j0 + tx] = accT[t] + sB[t];

  if (ti != tj) {                         // mirror tile via LDS transpose
    __syncthreads();
#pragma unroll
    for (int t = 0; t < NT_; ++t) sT[t][ty][tx] = accT[t] + sB[t];
    __syncthreads();
#pragma unroll
    for (int t = 0; t < NT_; ++t)
      out[ob + (size_t)t * WW_ + (size_t)(j0 + ty) * W_ + i0 + tx] = sT[t][tx][ty];
  }
}

// ---------------------------------------------------------------------------
extern "C" void kernel_launch(void* const* d_in, const int* in_sizes, int n_in,
                              void* d_out, int out_size, void* d_ws, size_t ws_size,
                              hipStream_t stream) {
  const float* edge  = (const float*)d_in[0];
  const float* node  = (const float*)d_in[1];
  const float* w_aff = (const float*)d_in[2];
  const float* gamma = (const float*)d_in[3];
  const float* beta  = (const float*)d_in[4];
  const float* w_out = (const float*)d_in[5];
  const float* b_out = (const float*)d_in[6];
  float* out = (float*)d_out;

  float* ws   = (float*)d_ws;             // needs ~136 MB of scratch
  float* x    = ws + X_OFF;
  float* nf   = ws + NF_OFF;
  float* ng   = ws + NG_OFF;
  float* psum = ws + PSUM_OFF;
  float* psq  = ws + PSQ_OFF;
  float* scsh = ws + SCSH_OFF;

  k1_nodeproj<<<dim3(512), dim3(256), 0, stream>>>(node, w_aff, nf, ng);
  k2_edgeproj<<<dim3(NPART_), dim3(256), 0, stream>>>(edge, w_aff, nf, ng, x, psum, psq);
  k3_stats<<<dim3(1), dim3(64), 0, stream>>>(psum, psq, gamma, beta, scsh);
  k4_out<<<dim3(136, B_), dim3(256), 0, stream>>>(x, w_out, b_out, scsh, out);
}